// AttentionDecoder_22471268893342
// MI455X (gfx1250) — compile-verified
//
#include <hip/hip_runtime.h>
#include <cstdint>

#define H_ 1024
#define V_ 50257
#define L_ 2048

typedef __attribute__((ext_vector_type(2))) float v2f;
typedef __attribute__((ext_vector_type(8))) float v8f;

__device__ __forceinline__ float sigmf_(float x) { return 1.0f / (1.0f + expf(-x)); }

__device__ __forceinline__ float waveMax_(float v) {
  for (int o = 16; o > 0; o >>= 1) v = fmaxf(v, __shfl_xor(v, o, 32));
  return v;
}
__device__ __forceinline__ float waveSum_(float v) {
  for (int o = 16; o > 0; o >>= 1) v += __shfl_xor(v, o, 32);
  return v;
}

// ---------------------------------------------------------------------------
// Embedding gather + build xh = [embed | h0_fwd] and comb[0:H] = embed
// ---------------------------------------------------------------------------
__global__ __launch_bounds__(256) void k_prep(const long long* __restrict__ xt,
                                              const float* __restrict__ h0,
                                              const float* __restrict__ emb,
                                              float* __restrict__ xh,
                                              float* __restrict__ comb) {
  const int i = blockIdx.x * blockDim.x + threadIdx.x;
  const long long tok = xt[0];
  if (i < H_) {
    const float e = emb[(size_t)tok * H_ + i];
    xh[i] = e;
    xh[H_ + i] = h0[i];
    comb[i] = e;
  }
}

// ---------------------------------------------------------------------------
// GEMV via V_WMMA_F32_16X16X4_F32.
//   out[r] (+)= dot(W[r,0:K], x[0:K]) [+ b1[r]] [+ b2[r]] ; optional relu.
// One wave computes 16 rows (A = 16x4 f32 tile), vector lives in B column N=0.
// A layout (ISA 7.12.2, 32-bit A 16x4): lanes 0-15 -> K=0,1 ; lanes 16-31 ->
// K=2,3 ; so each lane loads 8B contiguous from its row.  B mirrors this:
// VGPR0/1 hold rows K={0,2}/{1,3}; only N=0 lanes (0 and 16) carry data.
// D column 0: lane 0 VGPR i = row r0+i, lane 16 VGPR i = row r0+8+i.
// Vector x is staged to LDS with CDNA5 async global->LDS copies.
// Requires K % 1024 == 0 (true here: K = 1024/2048/3072) so the async staging
// loop is divergence-free and WMMA sees EXEC = all ones.
// ---------------------------------------------------------------------------
__global__ __launch_bounds__(256) void k_gemv_wmma(const float* __restrict__ W,
                                                   const float* __restrict__ x,
                                                   const float* __restrict__ b1,
                                                   const float* __restrict__ b2,
                                                   float* __restrict__ out,
                                                   int M, int K,
                                                   int acc_flag, int relu_flag) {
  __shared__ float sx[3072];
  const int t = threadIdx.x;

  // --- async stage x -> LDS (16B per transaction) --------------------------
  {
    const int nTxn = K >> 2;  // multiples of 256 for all K used here
    const unsigned ldsBase = (unsigned)(size_t)(&sx[0]);
    const unsigned long long gBase = (unsigned long long)(const void*)x;
    for (int i = t; i < nTxn; i += 256) {
      unsigned dst = ldsBase + (unsigned)(i << 4);
      unsigned long long src = gBase + ((unsigned long long)i << 4);
      asm volatile("global_load_async_to_lds_b128 %0, %1, off"
                   :: "v"(dst), "v"(src) : "memory");
    }
    asm volatile("s_wait_asynccnt 0x0" ::: "memory");
    __syncthreads();
  }

  const int lane = t & 31;
  const int wv = t >> 5;
  const int m = lane & 15;
  const int kh = lane >> 4;          // 0 -> K 0,1 ; 1 -> K 2,3
  const int r0 = blockIdx.x * 128 + wv * 16;
  const int row = r0 + m;
  const bool rowOK = row < M;
  const float* wrow = W + (size_t)(rowOK ? row : (M - 1)) * K + 2 * kh;
  const float* sxp = sx + 2 * kh;
  const bool isN0 = (m == 0);

  v8f acc = {0.f, 0.f, 0.f, 0.f, 0.f, 0.f, 0.f, 0.f};
#pragma unroll 4
  for (int k = 0; k < K; k += 4) {
    v2f a = *(const v2f*)(wrow + k);        // 8B contiguous weight load
    if (!rowOK) { a.x = 0.f; a.y = 0.f; }
    v2f b = *(const v2f*)(sxp + k);         // LDS broadcast of vector slice
    if (!isN0) { b.x = 0.f; b.y = 0.f; }    // vector occupies column N=0 only
    acc = __builtin_amdgcn_wmma_f32_16x16x4_f32(false, a, false, b,
                                                (short)0, acc, false, false);
  }

  if (isN0) {                                // lanes 0 and 16 hold column N=0
    const int rbase = r0 + kh * 8;
    for (int i = 0; i < 8; ++i) {
      const int r = rbase + i;
      if (r < M) {
        float v = acc[i];
        if (b1) v += b1[r];
        if (b2) v += b2[r];
        if (acc_flag) v += out[r];
        if (relu_flag) v = fmaxf(v, 0.f);
        out[r] = v;
      }
    }
  }
}

// ---------------------------------------------------------------------------
// Softmax over n elements (single block)
// ---------------------------------------------------------------------------
__global__ __launch_bounds__(1024) void k_softmax(const float* __restrict__ s,
                                                  float* __restrict__ w, int n) {
  __shared__ float red[32];
  __shared__ float bc[2];
  const int t = threadIdx.x;
  const int lane = t & 31, wv = t >> 5;
  float m = -3.4e38f;
  for (int i = t; i < n; i += blockDim.x) m = fmaxf(m, s[i]);
  m = waveMax_(m);
  if (lane == 0) red[wv] = m;
  __syncthreads();
  if (wv == 0) {
    float v = (lane < (int)(blockDim.x >> 5)) ? red[lane] : -3.4e38f;
    v = waveMax_(v);
    if (lane == 0) bc[0] = v;
  }
  __syncthreads();
  m = bc[0];
  float sum = 0.f;
  for (int i = t; i < n; i += blockDim.x) sum += expf(s[i] - m);
  sum = waveSum_(sum);
  if (lane == 0) red[wv] = sum;
  __syncthreads();
  if (wv == 0) {
    float v = (lane < (int)(blockDim.x >> 5)) ? red[lane] : 0.f;
    v = waveSum_(v);
    if (lane == 0) bc[1] = v;
  }
  __syncthreads();
  const float inv = 1.0f / bc[1];
  for (int i = t; i < n; i += blockDim.x) w[i] = expf(s[i] - m) * inv;
}

// ---------------------------------------------------------------------------
// Log-softmax over n elements (single block)
// ---------------------------------------------------------------------------
__global__ __launch_bounds__(1024) void k_logsoftmax(const float* __restrict__ s,
                                                     float* __restrict__ o, int n) {
  __shared__ float red[32];
  __shared__ float bc[2];
  const int t = threadIdx.x;
  const int lane = t & 31, wv = t >> 5;
  float m = -3.4e38f;
  for (int i = t; i < n; i += blockDim.x) m = fmaxf(m, s[i]);
  m = waveMax_(m);
  if (lane == 0) red[wv] = m;
  __syncthreads();
  if (wv == 0) {
    float v = (lane < (int)(blockDim.x >> 5)) ? red[lane] : -3.4e38f;
    v = waveMax_(v);
    if (lane == 0) bc[0] = v;
  }
  __syncthreads();
  m = bc[0];
  float sum = 0.f;
  for (int i = t; i < n; i += blockDim.x) sum += expf(s[i] - m);
  sum = waveSum_(sum);
  if (lane == 0) red[wv] = sum;
  __syncthreads();
  if (wv == 0) {
    float v = (lane < (int)(blockDim.x >> 5)) ? red[lane] : 0.f;
    v = waveSum_(v);
    if (lane == 0) bc[1] = v;
  }
  __syncthreads();
  const float lse = m + logf(bc[1]);
  for (int i = t; i < n; i += blockDim.x) o[i] = s[i] - lse;
}

// ---------------------------------------------------------------------------
// att_applied = att_weight @ encoder_outputs  (E row-major [L, 2H])
// Deterministic two-stage column reduction (no float atomics).
// ---------------------------------------------------------------------------
__global__ __launch_bounds__(256) void k_att_partial(const float* __restrict__ aw,
                                                     const float* __restrict__ E,
                                                     float* __restrict__ part) {
  const int j = blockIdx.x * 256 + threadIdx.x;   // 0..2047 (coalesced in j)
  const int lc = blockIdx.y;                      // 0..7
  float acc = 0.f;
  const int l0 = lc * 256;
  for (int l = l0; l < l0 + 256; ++l)
    acc = fmaf(aw[l], E[(size_t)l * (2 * H_) + j], acc);
  part[lc * (2 * H_) + j] = acc;
}

__global__ __launch_bounds__(256) void k_att_reduce(const float* __restrict__ part,
                                                    float* __restrict__ dst) {
  const int j = blockIdx.x * 256 + threadIdx.x;
  float s = 0.f;
  for (int lc = 0; lc < 8; ++lc) s += part[lc * (2 * H_) + j];
  dst[j] = s;
}

// ---------------------------------------------------------------------------
// LSTM pointwise update for one direction (gate order i, f, g, o)
// ---------------------------------------------------------------------------
__global__ __launch_bounds__(256) void k_lstm(const float* __restrict__ g,
                                              const float* __restrict__ c0,
                                              float* __restrict__ hWs,
                                              float* __restrict__ hOut,
                                              float* __restrict__ cOut) {
  const int e = blockIdx.x * 256 + threadIdx.x;
  if (e < H_) {
    const float ig = sigmf_(g[e]);
    const float fg = sigmf_(g[H_ + e]);
    const float gg = tanhf(g[2 * H_ + e]);
    const float og = sigmf_(g[3 * H_ + e]);
    const float c2 = fg * c0[e] + ig * gg;
    const float h2 = og * tanhf(c2);
    hWs[e] = h2;
    hOut[e] = h2;
    cOut[e] = c2;
  }
}

// ---------------------------------------------------------------------------
extern "C" void kernel_launch(void* const* d_in, const int* in_sizes, int n_in,
                              void* d_out, int out_size, void* d_ws, size_t ws_size,
                              hipStream_t stream) {
  (void)in_sizes; (void)n_in; (void)out_size; (void)ws_size;

  const long long* x  = (const long long*)d_in[0];
  const float* h0     = (const float*)d_in[1];
  const float* c0     = (const float*)d_in[2];
  const float* enc    = (const float*)d_in[3];
  const float* emb    = (const float*)d_in[4];
  const float* attn_w = (const float*)d_in[5];
  const float* attn_b = (const float*)d_in[6];
  const float* comb_w = (const float*)d_in[7];
  const float* comb_b = (const float*)d_in[8];
  const float* w_ih_f = (const float*)d_in[9];
  const float* w_hh_f = (const float*)d_in[10];
  const float* b_ih_f = (const float*)d_in[11];
  const float* b_hh_f = (const float*)d_in[12];
  const float* w_ih_b = (const float*)d_in[13];
  const float* w_hh_b = (const float*)d_in[14];
  const float* b_ih_b = (const float*)d_in[15];
  const float* b_hh_b = (const float*)d_in[16];
  const float* out_w  = (const float*)d_in[17];
  const float* out_b  = (const float*)d_in[18];

  float* ws       = (float*)d_ws;
  float* xh       = ws + 0;       // 2048   [embed | h0_fwd]
  float* score    = ws + 2048;    // 2048
  float* comb     = ws + 4096;    // 3072   [embed | att_applied]
  float* lstm_in  = ws + 7168;    // 1024
  float* gates_f  = ws + 8192;    // 4096
  float* gates_b  = ws + 12288;   // 4096
  float* lstm_out = ws + 16384;   // 2048   [hf | hb]
  float* logits   = ws + 18432;   // 50257
  float* part     = ws + 68864;   // 8 * 2048 partials

  float* o    = (float*)d_out;
  float* logp = o;                 // [V]
  float* hn   = o + V_;            // [2H]  hf | hb
  float* cn   = o + V_ + 2048;     // [2H]  cf | cb
  float* attw = o + V_ + 4096;     // [L]

  // 1. embed gather, xh, comb[0:H]
  k_prep<<<4, 256, 0, stream>>>(x, h0, emb, xh, comb);
  // 2. attention scores: [L] = attn_w[L,2H] @ xh + attn_b
  k_gemv_wmma<<<L_ / 128, 256, 0, stream>>>(attn_w, xh, attn_b, nullptr,
                                            score, L_, 2 * H_, 0, 0);
  // 3. softmax -> att_weight (final output + reused below)
  k_softmax<<<1, 1024, 0, stream>>>(score, attw, L_);
  // 4. att_applied = attw @ enc  ->  comb[H:3H]
  k_att_partial<<<dim3(8, 8), 256, 0, stream>>>(attw, enc, part);
  k_att_reduce<<<8, 256, 0, stream>>>(part, comb + H_);
  // 5. lstm_in = relu(comb_w[H,3H] @ comb + comb_b)
  k_gemv_wmma<<<H_ / 128, 256, 0, stream>>>(comb_w, comb, comb_b, nullptr,
                                            lstm_in, H_, 3 * H_, 0, 1);
  // 6. gate pre-activations, both directions
  k_gemv_wmma<<<32, 256, 0, stream>>>(w_ih_f, lstm_in, b_ih_f, b_hh_f,
                                      gates_f, 4 * H_, H_, 0, 0);
  k_gemv_wmma<<<32, 256, 0, stream>>>(w_hh_f, h0, nullptr, nullptr,
                                      gates_f, 4 * H_, H_, 1, 0);
  k_gemv_wmma<<<32, 256, 0, stream>>>(w_ih_b, lstm_in, b_ih_b, b_hh_b,
                                      gates_b, 4 * H_, H_, 0, 0);
  k_gemv_wmma<<<32, 256, 0, stream>>>(w_hh_b, h0 + H_, nullptr, nullptr,
                                      gates_b, 4 * H_, H_, 1, 0);
  // 7. pointwise LSTM update (h/c outputs + lstm_out)
  k_lstm<<<4, 256, 0, stream>>>(gates_f, c0, lstm_out, hn, cn);
  k_lstm<<<4, 256, 0, stream>>>(gates_b, c0 + H_, lstm_out + H_, hn + H_, cn + H_);
  // 8. logits = out_w[V,2H] @ lstm_out + out_b   (the 412 MB stream)
  k_gemv_wmma<<<(V_ + 127) / 128, 256, 0, stream>>>(out_w, lstm_out, out_b, nullptr,
                                                    logits, V_, 2 * H_, 0, 0);
  // 9. log-softmax -> logp
  k_logsoftmax<<<1, 1024, 0, stream>>>(logits, logp, V_);
}